// TMPHN_11974368821733
// MI455X (gfx1250) — compile-verified
//
#include <hip/hip_runtime.h>
#include <math.h>

#define N_NODES 50000
#define FEAT 5
#define HID 128
#define MNEIG 32
#define BSZ 8
#define NCLS 3
#define AGG_STRIDE 132   // 128 + 4 pad -> conflict-free WMMA A-fragment reads

typedef __attribute__((ext_vector_type(2))) float v2f;
typedef __attribute__((ext_vector_type(8))) float v8f;

// ---------------- Layer 1: mean-aggregate X (5 feats) -> relu(agg @ W1) -----
__global__ void tmphn_layer1(const float* __restrict__ X,
                             const float* __restrict__ W1,
                             const int* __restrict__ neig,
                             float* __restrict__ H1) {
    __shared__ float agg[FEAT];
    __shared__ float w1s[FEAT * HID];
    const int t = threadIdx.x;          // 128 threads
    const int node = blockIdx.x;

    for (int s = t; s < FEAT * HID; s += HID) w1s[s] = W1[s];
    if (t < FEAT) agg[t] = 0.0f;
    __syncthreads();

    if (t < MNEIG) {
        const int nb = neig[node * MNEIG + t];
        #pragma unroll
        for (int k = 0; k < FEAT; ++k)
            atomicAdd(&agg[k], X[(long)nb * FEAT + k] * (1.0f / MNEIG));
    }
    __syncthreads();

    float s = 0.0f;
    #pragma unroll
    for (int k = 0; k < FEAT; ++k) s += agg[k] * w1s[k * HID + t];
    H1[(long)node * HID + t] = s > 0.0f ? s : 0.0f;
}

// ---------------- Layer 2: mean-aggregate H1 -> relu(agg @ W2) via f32 WMMA -
// Block = 256 threads (8 waves), 16 nodes per block. Wave w owns N-tile w.
__global__ void tmphn_layer2(const float* __restrict__ H1,
                             const float* __restrict__ W2,
                             const int* __restrict__ neig,
                             float* __restrict__ H2) {
    __shared__ float agg[16][AGG_STRIDE];
    __shared__ int nbr[16][MNEIG];
    const int t = threadIdx.x;               // 0..255
    const int node0 = blockIdx.x * 16;

    for (int s = t; s < 16 * MNEIG; s += 256)
        nbr[s / MNEIG][s % MNEIG] = neig[(long)(node0 + s / MNEIG) * MNEIG + s % MNEIG];
    for (int s = t; s < 16 * AGG_STRIDE; s += 256)
        (&agg[0][0])[s] = 0.0f;
    __syncthreads();

    // Aggregation: thread (c = t&127, h = t>>7) sums half the neighbors of each node.
    const int c = t & (HID - 1);
    const int h = t >> 7;                    // 0 or 1
    for (int i = 0; i < 16; ++i) {
        float s = 0.0f;
        #pragma unroll
        for (int j = 0; j < 16; ++j) {
            const int nb = nbr[i][h * 16 + j];
            s += H1[(long)nb * HID + c];
        }
        atomicAdd(&agg[i][c], s * (1.0f / MNEIG));   // ds_add_f32
    }
    __syncthreads();

    // GEMM [16x128] @ [128x128]: wave w -> output columns [16w, 16w+16).
    const int wave = t >> 5;
    const int lane = t & 31;
    const int n0 = wave * 16;
    const int lm = lane & 15;                // M (lanes<16) / shares N index
    const int hi = lane >> 4;                // K-half selector
    v8f acc = {};
    for (int k = 0; k < HID; k += 4) {
        const int ka = k + hi * 2;           // lanes 0-15: K=k,k+1 ; lanes 16-31: K=k+2,k+3
        v2f a = *(const v2f*)&agg[lm][ka];   // 8B aligned: AGG_STRIDE even, ka even
        v2f b;
        b.x = W2[(long)ka * HID + n0 + lm];
        b.y = W2[(long)(ka + 1) * HID + n0 + lm];
        acc = __builtin_amdgcn_wmma_f32_16x16x4_f32(
            /*neg_a=*/false, a, /*neg_b=*/false, b,
            /*c_mod=*/(short)0, acc, /*reuse_a=*/false, /*reuse_b=*/false);
    }
    // D layout: VGPR r holds M=r (lanes 0-15) and M=r+8 (lanes 16-31), N = n0+lm.
    #pragma unroll
    for (int r = 0; r < 8; ++r) {
        const int m = r + hi * 8;
        const float v = acc[r];
        H2[(long)(node0 + m) * HID + n0 + lm] = v > 0.0f ? v : 0.0f;
    }
}

// ---------------- Head: per-batch sum of gathered H2 rows -------------------
__global__ void tmphn_zero_bsum(float* __restrict__ bsum) {
    bsum[threadIdx.x] = 0.0f;                // 1024 threads = BSZ*HID
}

#define CHUNK 500                            // 100 chunks * 500 = 50000
__global__ void tmphn_head_sum(const float* __restrict__ H2,
                               const int* __restrict__ ts,
                               float* __restrict__ bsum) {
    const int c = threadIdx.x;               // 128 threads: one column each
    const int b = blockIdx.y;
    const int start = blockIdx.x * CHUNK;
    float acc = 0.0f;
    for (int i = 0; i < CHUNK; ++i) {
        const int idx = ts[(long)b * N_NODES + start + i];
        acc += H2[(long)idx * HID + c];
    }
    atomicAdd(&bsum[b * HID + c], acc);
}

// mean commutes with the linear head: out = logsoftmax((sum/N) @ W + b)
__global__ void tmphn_head_final(const float* __restrict__ bsum,
                                 const float* __restrict__ W,
                                 const float* __restrict__ bias,
                                 float* __restrict__ out) {
    const int b = threadIdx.x;
    if (b >= BSZ) return;
    float logit[NCLS];
    #pragma unroll
    for (int c2 = 0; c2 < NCLS; ++c2) {
        float s = bias[c2];
        for (int k = 0; k < HID; ++k)
            s += (bsum[b * HID + k] * (1.0f / N_NODES)) * W[k * NCLS + c2];
        logit[c2] = s;
    }
    float mx = fmaxf(logit[0], fmaxf(logit[1], logit[2]));
    float se = expf(logit[0] - mx) + expf(logit[1] - mx) + expf(logit[2] - mx);
    float lse = mx + logf(se);
    #pragma unroll
    for (int c2 = 0; c2 < NCLS; ++c2) out[b * NCLS + c2] = logit[c2] - lse;
}

extern "C" void kernel_launch(void* const* d_in, const int* in_sizes, int n_in,
                              void* d_out, int out_size, void* d_ws, size_t ws_size,
                              hipStream_t stream) {
    const float* X    = (const float*)d_in[0];
    const float* W1   = (const float*)d_in[1];
    const float* W2   = (const float*)d_in[2];
    const float* W    = (const float*)d_in[3];
    const float* bias = (const float*)d_in[4];
    const int*   neig = (const int*)d_in[5];
    const int*   ts   = (const int*)d_in[6];
    float* out = (float*)d_out;

    float* H1   = (float*)d_ws;                       // 50000*128 f32 = 25.6 MB
    float* H2   = H1 + (size_t)N_NODES * HID;         // 25.6 MB
    float* bsum = H2 + (size_t)N_NODES * HID;         // 8*128 f32

    tmphn_layer1<<<N_NODES, HID, 0, stream>>>(X, W1, neig, H1);
    tmphn_layer2<<<N_NODES / 16, 256, 0, stream>>>(H1, W2, neig, H2);
    tmphn_zero_bsum<<<1, BSZ * HID, 0, stream>>>(bsum);
    tmphn_head_sum<<<dim3(N_NODES / CHUNK, BSZ), HID, 0, stream>>>(H2, ts, bsum);
    tmphn_head_final<<<1, 32, 0, stream>>>(bsum, W, bias, out);
}